// Dct2_60043642798468
// MI455X (gfx1250) — compile-verified
//
#include <hip/hip_runtime.h>

// 2-D DCT-II of 4096x4096 fp32 as two dense GEMMs against the cosine basis:
//   T   = C * x          (GEMM1: B = x, row-major [k][n])
//   out = T * C^T        (GEMM2: B = C^T, i.e. row-major reads of C -> [n][k])
// Native fp32 matrix op V_WMMA_F32_16X16X4_F32, double-buffered LDS tiles fed
// by GLOBAL_LOAD_ASYNC_TO_LDS_B128 (ASYNCcnt) when the toolchain exposes it.

typedef __attribute__((ext_vector_type(2))) float v2f;
typedef __attribute__((ext_vector_type(8))) float v8f;
typedef int i32x4 __attribute__((vector_size(16)));   // matches builtin proto

#define GLOBAL_AS __attribute__((address_space(1)))
#define LDS_AS    __attribute__((address_space(3)))

#if defined(__HIP_DEVICE_COMPILE__) &&                                   \
    __has_builtin(__builtin_amdgcn_global_load_async_to_lds_b128) &&     \
    __has_builtin(__builtin_amdgcn_s_wait_asynccnt)
#define USE_ASYNC_LDS 1
#else
#define USE_ASYNC_LDS 0
#endif

constexpr int NSZ = 4096;
constexpr int BM  = 128;   // workgroup M tile
constexpr int BN  = 256;   // workgroup N tile
constexpr int BK  = 32;    // k-chunk
constexpr int NCH = NSZ / BK;   // 128 k-chunks

// 16-byte global -> LDS copy: async (no VGPR data path) or sync fallback.
__device__ __forceinline__ void copy16(const float* __restrict__ g,
                                       float* __restrict__ l) {
#if USE_ASYNC_LDS
  __builtin_amdgcn_global_load_async_to_lds_b128(
      (GLOBAL_AS i32x4*)g, (LDS_AS i32x4*)l, /*offset=*/0, /*cpol=*/0);
#else
  *(float4*)l = *(const float4*)g;
#endif
}

// ---------------------------------------------------------------------------
// C[u,i] = cos(pi*(2i+1)*u / (2N)); phase reduced mod 2*pi exactly in ints.
// ---------------------------------------------------------------------------
__global__ __launch_bounds__(256) void dct_basis_kernel(float* __restrict__ Cmat) {
  const int idx = blockIdx.x * 256 + threadIdx.x;
  const int u = idx >> 12;
  const int i = idx & (NSZ - 1);
  const unsigned ph = ((unsigned)(2 * i + 1) * (unsigned)u) & (4u * NSZ - 1u);
  const float w = 3.14159265358979323846f / (2.0f * (float)NSZ);
  Cmat[idx] = cosf((float)ph * w);
}

// ---------------------------------------------------------------------------
// D = A * B (row-major 4096^2). TRANSB=1: logical B[k][n] = Bmem[n][k].
// 256 threads = 8 waves (2x4); WG tile 128x256; wave tile 64x64 = 4x4 WMMA tiles.
// ---------------------------------------------------------------------------
template <bool TRANSB>
__global__ __launch_bounds__(256)
void dct_gemm_wmma(const float* __restrict__ A, const float* __restrict__ B,
                   float* __restrict__ D) {
  constexpr int LDA = 36;                        // A LDS row stride (floats)
  constexpr int LDB = TRANSB ? 36 : (BN + 8);    // 36 ([n][k]) or 264 ([k][n])
  constexpr int AFL = BM * LDA;                  // 4608 floats
  constexpr int BFL = TRANSB ? BN * LDB : BK * LDB;  // 9216 / 8448 floats
  constexpr int BUF = AFL + BFL;                 // one k-chunk buffer

  extern __shared__ float lds[];                 // 2 * BUF floats (dynamic)

  const int tid  = threadIdx.x;
  const int bm   = blockIdx.y * BM;
  const int bn   = blockIdx.x * BN;
  const int lane = tid & 31;
  const int lr   = lane & 15;   // M (A/D) or N (B/D) index in 16x16 tile
  const int lh   = lane >> 4;   // half-wave: K-pair select / M+8 row select
  const int wv   = tid >> 5;
  const int wm   = (wv >> 2) * 64;
  const int wn   = (wv & 3) * 64;

  v8f acc[4][4];
#pragma unroll
  for (int i = 0; i < 4; ++i)
#pragma unroll
    for (int j = 0; j < 4; ++j)
#pragma unroll
      for (int p = 0; p < 8; ++p) acc[i][j][p] = 0.0f;

  // ---- stage one k-chunk into buf: 12 x 16B per thread (A:4, B:8) ----
  auto stage = [&](int k0, float* __restrict__ buf) {
    float* la = buf;
    float* lb = buf + AFL;
#pragma unroll
    for (int it = 0; it < 4; ++it) {             // A: lds_a[r][c] = A[bm+r][k0+c]
      const int idx = tid + it * 256;
      const int r = idx >> 3, c4 = (idx & 7) * 4;
      copy16(&A[(size_t)(bm + r) * NSZ + (k0 + c4)], &la[r * LDA + c4]);
    }
    if (TRANSB) {                                // lds_b[n][c] = Bmem[bn+n][k0+c]
#pragma unroll
      for (int it = 0; it < 8; ++it) {
        const int idx = tid + it * 256;
        const int n = idx >> 3, c4 = (idx & 7) * 4;
        copy16(&B[(size_t)(bn + n) * NSZ + (k0 + c4)], &lb[n * LDB + c4]);
      }
    } else {                                     // lds_b[k][n] = B[k0+k][bn+n]
#pragma unroll
      for (int it = 0; it < 8; ++it) {
        const int idx = tid + it * 256;
        const int kr = idx >> 6, n4 = (idx & 63) * 4;
        copy16(&B[(size_t)(k0 + kr) * NSZ + (bn + n4)], &lb[kr * LDB + n4]);
      }
    }
  };

  // ---- 8 k-steps x 16 tiles = 128 v_wmma_f32_16x16x4_f32 per chunk ----
  auto compute = [&](const float* __restrict__ buf) {
    const float* la = buf;
    const float* lb = buf + AFL;
#pragma unroll
    for (int kk = 0; kk < BK; kk += 4) {
      v2f af[4], bf[4];
#pragma unroll
      for (int i = 0; i < 4; ++i)   // A frag: lane=M, VGPR pair = K (kk+2lh,+1)
        af[i] = *(const v2f*)&la[(wm + i * 16 + lr) * LDA + kk + 2 * lh];
#pragma unroll
      for (int j = 0; j < 4; ++j) {
        if (TRANSB) {               // [n][k]: consecutive-K pair, one b64
          bf[j] = *(const v2f*)&lb[(wn + j * 16 + lr) * LDB + kk + 2 * lh];
        } else {                    // [k][n]: two b32, rows kk+2lh, kk+2lh+1
          v2f t;
          t[0] = lb[(kk + 2 * lh + 0) * LDB + (wn + j * 16 + lr)];
          t[1] = lb[(kk + 2 * lh + 1) * LDB + (wn + j * 16 + lr)];
          bf[j] = t;
        }
      }
#pragma unroll
      for (int i = 0; i < 4; ++i)
#pragma unroll
        for (int j = 0; j < 4; ++j)
          acc[i][j] = __builtin_amdgcn_wmma_f32_16x16x4_f32(
              false, af[i], false, bf[j], (short)0, acc[i][j], false, false);
    }
  };

  // ---- double-buffered main loop ----
  stage(0, lds);
  for (int c = 0; c < NCH; ++c) {
    if (c + 1 < NCH) {
      stage((c + 1) * BK, &lds[((c + 1) & 1) * BUF]);
#if USE_ASYNC_LDS
      __builtin_amdgcn_s_wait_asynccnt(12);   // chunk c's 12 ops retired
#endif
    } else {
#if USE_ASYNC_LDS
      __builtin_amdgcn_s_wait_asynccnt(0);
#endif
    }
    __syncthreads();
    compute(&lds[(c & 1) * BUF]);
    __syncthreads();
  }

  // ---- epilogue: D element (row = p + 8*lh, col = lr) per accumulator VGPR p ----
#pragma unroll
  for (int i = 0; i < 4; ++i)
#pragma unroll
    for (int j = 0; j < 4; ++j) {
      const int col = bn + wn + j * 16 + lr;
#pragma unroll
      for (int p = 0; p < 8; ++p) {
        const int row = bm + wm + i * 16 + p + 8 * lh;
        D[(size_t)row * NSZ + col] = acc[i][j][p];
      }
    }
}

// ---------------------------------------------------------------------------
static constexpr size_t gemm_shbytes(bool transb) {
  const int AFL = BM * 36;
  const int BFL = transb ? BN * 36 : BK * (BN + 8);
  return (size_t)2 * (AFL + BFL) * sizeof(float);
}

extern "C" void kernel_launch(void* const* d_in, const int* in_sizes, int n_in,
                              void* d_out, int out_size, void* d_ws, size_t ws_size,
                              hipStream_t stream) {
  (void)in_sizes; (void)n_in; (void)out_size; (void)ws_size;
  const float* x   = (const float*)d_in[0];
  float*       out = (float*)d_out;
  float*       Cm  = (float*)d_ws;                 // 64 MB cosine basis
  float*       T   = Cm + (size_t)NSZ * NSZ;       // 64 MB intermediate

  dct_basis_kernel<<<(NSZ * NSZ) / 256, 256, 0, stream>>>(Cm);

  const dim3 grid(NSZ / BN, NSZ / BM);   // 16 x 32 workgroups
  const dim3 block(256);

  // T = C * x
  dct_gemm_wmma<false><<<grid, block, gemm_shbytes(false), stream>>>(Cm, x, T);
  // out = T * C^T   (C^T read as row-major C)
  dct_gemm_wmma<true><<<grid, block, gemm_shbytes(true), stream>>>(T, Cm, out);
}